// SelfAttention_72292889526979
// MI455X (gfx1250) — compile-verified
//
#include <hip/hip_runtime.h>

typedef __attribute__((ext_vector_type(2))) float v2f;
typedef __attribute__((ext_vector_type(8))) float v8f;

#define BM 128
#define BN 128
#define BK 16
#define LDSK 20      // padded row stride (floats): 80B rows (16B aligned), conflict-free frags
#define KDIM 1024
#define NDIM 1024

// One 16-byte global -> LDS async copy (ASYNCcnt-tracked, no VGPR staging).
// Generic shared-pointer low 32 bits == LDS byte offset (ISA 10.2).
__device__ __forceinline__ void async_cp16(const float* gptr, const float* lptr)
{
    unsigned int       l = (unsigned int)(unsigned long long)lptr;
    unsigned long long g = (unsigned long long)gptr;
    asm volatile("global_load_async_to_lds_b128 %0, %1, off"
                 :: "v"(l), "v"(g) : "memory");
}

__device__ __forceinline__ void wait_async0()
{
    asm volatile("s_wait_asynccnt 0x0" ::: "memory");
}

// C[M x 1024] = A[M x 1024] @ W^T + bias, W stored (N,K) row-major.
// qkvMap==1: output column c reads w_in row (c/64)*192 + 128 + (c%64)  (per-head "v" chunk).
__global__ __launch_bounds__(256) void gemm_bias_wmma(
    const float* __restrict__ A, const float* __restrict__ W,
    const float* __restrict__ bias, float* __restrict__ C, int qkvMap)
{
    __shared__ float As[2][BM * LDSK];
    __shared__ float Bs[2][BN * LDSK];

    const int tid   = threadIdx.x;
    const int tileN = blockIdx.x * BN;
    const int tileM = blockIdx.y * BM;

    const int wave = tid >> 5;
    const int lane = tid & 31;
    const int mSub = (wave & 3) * 32;   // 4 M-subtiles of 32
    const int nSub = (wave >> 2) * 64;  // 2 N-strips of 64
    const int rowL = lane & 15;
    const int hi   = lane >> 4;         // selects K pair / M half per f32 operand layout

    // ---- per-thread staging coordinates: tiles are 128 rows x 16 floats = 512 x 16B chunks,
    //      2 chunks per thread; chunk i: idx = tid + i*256, row = idx>>2, col = (idx&3)*4 ----
    int arows[2], bcol[2];
    const float* arow[2];
    const float* brow[2];
    #pragma unroll
    for (int i = 0; i < 2; ++i) {
        int idx = tid + i * 256;
        int r   = idx >> 2;              // 0..127
        int c4  = (idx & 3) << 2;        // 0,4,8,12
        arows[i] = r; bcol[i] = c4;
        arow[i] = A + (size_t)(tileM + r) * KDIM + c4;
        int n   = tileN + r;
        int g   = qkvMap ? ((n >> 6) * 192 + 128 + (n & 63)) : n;
        brow[i] = W + (size_t)g * KDIM + c4;
    }

    v8f acc[8] = {};   // acc[mh*4 + j]: mh = M half (0/1), j = N subtile (0..3)

    // ---- stage first tile (buffer 0) ----
    #pragma unroll
    for (int i = 0; i < 2; ++i) {
        async_cp16(arow[i], &As[0][arows[i] * LDSK + bcol[i]]);
        async_cp16(brow[i], &Bs[0][arows[i] * LDSK + bcol[i]]);
    }

    int buf = 0;
    for (int k0 = 0; k0 < KDIM; k0 += BK, buf ^= 1) {
        wait_async0();        // this thread's copies into `buf` complete
        __syncthreads();      // all threads' copies complete -> tile ready

        if (k0 + BK < KDIM) { // prefetch next tile into other buffer (all reads of it
                              // completed at the barrier that ended the previous iter)
            int nb = buf ^ 1;
            #pragma unroll
            for (int i = 0; i < 2; ++i) {
                async_cp16(arow[i] + k0 + BK, &As[nb][arows[i] * LDSK + bcol[i]]);
                async_cp16(brow[i] + k0 + BK, &Bs[nb][arows[i] * LDSK + bcol[i]]);
            }
        }

        const float* as = &As[buf][0];
        const float* bs = &Bs[buf][0];
        #pragma unroll
        for (int k = 0; k < BK; k += 4) {
            int kk = k + 2 * hi;  // lane's K pair per 16x4 f32 operand layout
            v2f a0 = *(const v2f*)(as + (mSub +  0 + rowL) * LDSK + kk);
            v2f a1 = *(const v2f*)(as + (mSub + 16 + rowL) * LDSK + kk);
            v2f b0 = *(const v2f*)(bs + (nSub +  0 + rowL) * LDSK + kk);
            v2f b1 = *(const v2f*)(bs + (nSub + 16 + rowL) * LDSK + kk);
            v2f b2 = *(const v2f*)(bs + (nSub + 32 + rowL) * LDSK + kk);
            v2f b3 = *(const v2f*)(bs + (nSub + 48 + rowL) * LDSK + kk);
            acc[0] = __builtin_amdgcn_wmma_f32_16x16x4_f32(false, a0, false, b0, (short)0, acc[0], false, false);
            acc[1] = __builtin_amdgcn_wmma_f32_16x16x4_f32(false, a0, false, b1, (short)0, acc[1], false, false);
            acc[2] = __builtin_amdgcn_wmma_f32_16x16x4_f32(false, a0, false, b2, (short)0, acc[2], false, false);
            acc[3] = __builtin_amdgcn_wmma_f32_16x16x4_f32(false, a0, false, b3, (short)0, acc[3], false, false);
            acc[4] = __builtin_amdgcn_wmma_f32_16x16x4_f32(false, a1, false, b0, (short)0, acc[4], false, false);
            acc[5] = __builtin_amdgcn_wmma_f32_16x16x4_f32(false, a1, false, b1, (short)0, acc[5], false, false);
            acc[6] = __builtin_amdgcn_wmma_f32_16x16x4_f32(false, a1, false, b2, (short)0, acc[6], false, false);
            acc[7] = __builtin_amdgcn_wmma_f32_16x16x4_f32(false, a1, false, b3, (short)0, acc[7], false, false);
        }
        __syncthreads();      // all waves done reading `buf` before it is re-filled
    }

    // ---- epilogue: lane owns column n = rowL(+16j); rows = mh*16 + hi*8 + r ----
    #pragma unroll
    for (int mh = 0; mh < 2; ++mh) {
        const int mBase = tileM + mSub + mh * 16 + hi * 8;
        #pragma unroll
        for (int j = 0; j < 4; ++j) {
            v8f a = acc[mh * 4 + j];
            int n = tileN + nSub + j * 16 + rowL;
            int bIdx = qkvMap ? ((n >> 6) * 192 + 128 + (n & 63)) : n;
            float bv = bias[bIdx];
            #pragma unroll
            for (int r = 0; r < 8; ++r) {
                C[(size_t)(mBase + r) * NDIM + n] = a[r] + bv;
            }
        }
    }
}

extern "C" void kernel_launch(void* const* d_in, const int* in_sizes, int n_in,
                              void* d_out, int out_size, void* d_ws, size_t ws_size,
                              hipStream_t stream)
{
    const float* x     = (const float*)d_in[0];
    const float* w_in  = (const float*)d_in[1];
    const float* b_in  = (const float*)d_in[2];
    const float* w_out = (const float*)d_in[3];
    const float* b_out = (const float*)d_in[4];
    float* out = (float*)d_out;
    float* V   = (float*)d_ws;   // intermediate (M x 1024) fp32

    const int M = in_sizes[0] / KDIM;  // b*l = 4096

    dim3 grid(NDIM / BN, M / BM);
    dim3 block(256);
    // softmax over m sums to 1 => attention output == v; network collapses to two GEMMs.
    gemm_bias_wmma<<<grid, block, 0, stream>>>(x, w_in, b_in, V, 1);
    gemm_bias_wmma<<<grid, block, 0, stream>>>(V, w_out, b_out, out, 0);
}